// QuadraticGNNLayer_33492154974253
// MI455X (gfx1250) — compile-verified
//
#include <hip/hip_runtime.h>

typedef float v2f __attribute__((ext_vector_type(2)));
typedef float v8f __attribute__((ext_vector_type(8)));

#define DD 128
#define LDSX 132        // padded LDS row stride (floats) -> conflict-free column reads
#define NEG_SLOPE 0.01f

// ---------------------------------------------------------------------------
// 16x16 output tile GEMM: out[row0:row0+16, n0:n0+16] = xt(16x128,in LDS) @ W + bias
// V_WMMA_F32_16X16X4_F32 (exact f32 math), K looped 128/4 = 32 steps.
// A frag (16x4 f32): lane m=lane&15 is row M; lanes 0-15 hold K=0,1 in v0,v1;
// lanes 16-31 hold K=2,3.  B frag (4x16): symmetric with N=lane&15.
// C/D (16x16 f32): VGPR r holds M=r (lanes 0-15) / M=r+8 (lanes 16-31), N=lane&15.
// ---------------------------------------------------------------------------
template <bool HAS_BIAS>
__device__ __forceinline__ void gemm_tile_store(
    const float* __restrict__ xt, const float* __restrict__ W,
    const float* __restrict__ bias, float* __restrict__ out,
    int row0, int n0, int lane, int nrows)
{
    const int m  = lane & 15;
    const int hi = lane >> 4;     // 0 or 1
    const int kb = hi * 2;        // K sub-offset for this half-wave
    const int nl = n0 + m;        // output column for B/C/D fragments

    const float bb = HAS_BIAS ? bias[nl] : 0.0f;

    v8f acc = {0.f, 0.f, 0.f, 0.f, 0.f, 0.f, 0.f, 0.f};
    #pragma unroll 8
    for (int k0 = 0; k0 < DD; k0 += 4) {
        v2f a, b;
        a.x = xt[m * LDSX + k0 + kb];
        a.y = xt[m * LDSX + k0 + kb + 1];
        b.x = W[(k0 + kb) * DD + nl];
        b.y = W[(k0 + kb + 1) * DD + nl];
        acc = __builtin_amdgcn_wmma_f32_16x16x4_f32(
            false, a, false, b, (short)0, acc, false, false);
    }

    float* op = out + (size_t)(row0 + hi * 8) * DD + nl;
    if (row0 + 16 <= nrows) {              // uniform fast path: no exec games
        #pragma unroll
        for (int r = 0; r < 8; ++r)
            op[(size_t)r * DD] = acc[r] + bb;
    } else {
        #pragma unroll
        for (int r = 0; r < 8; ++r)
            if (row0 + hi * 8 + r < nrows)
                op[(size_t)r * DD] = acc[r] + bb;
    }
}

// ---------------------------------------------------------------------------
// Kernel 1: fused node projections. One block = 16 rows of x; 8 waves each own
// a 16-wide column tile and produce k, q, v, skip for it.
// ---------------------------------------------------------------------------
__global__ __launch_bounds__(256) void node_proj_kernel(
    const float* __restrict__ x,
    const float* __restrict__ Wk, const float* __restrict__ bk,
    const float* __restrict__ Wq, const float* __restrict__ bq,
    const float* __restrict__ Wv, const float* __restrict__ bv,
    const float* __restrict__ Wskip,
    float* __restrict__ kout, float* __restrict__ qout,
    float* __restrict__ vout, float* __restrict__ sout, int nrows)
{
    __shared__ float xt[16 * LDSX];
    const int row0 = blockIdx.x * 16;
    const int tid  = threadIdx.x;

    // stage 16x128 f32 tile: 512 float4 loads across 256 threads
    if (row0 + 16 <= nrows) {              // uniform fast path
        #pragma unroll
        for (int i = tid; i < 512; i += 256) {
            const int r = (i * 4) / DD;
            const int c = (i * 4) % DD;
            const float4 val = *(const float4*)(x + (size_t)(row0 + r) * DD + c);
            float* p = &xt[r * LDSX + c];
            p[0] = val.x; p[1] = val.y; p[2] = val.z; p[3] = val.w;
        }
    } else {
        for (int i = tid; i < 512; i += 256) {
            const int r = (i * 4) / DD;
            const int c = (i * 4) % DD;
            float4 val = make_float4(0.f, 0.f, 0.f, 0.f);
            if (row0 + r < nrows)
                val = *(const float4*)(x + (size_t)(row0 + r) * DD + c);
            float* p = &xt[r * LDSX + c];
            p[0] = val.x; p[1] = val.y; p[2] = val.z; p[3] = val.w;
        }
    }
    __syncthreads();

    const int wave = tid >> 5;
    const int lane = tid & 31;
    const int n0   = wave * 16;

    gemm_tile_store<true >(xt, Wk,    bk, kout, row0, n0, lane, nrows);
    gemm_tile_store<true >(xt, Wq,    bq, qout, row0, n0, lane, nrows);
    gemm_tile_store<true >(xt, Wv,    bv, vout, row0, n0, lane, nrows);
    gemm_tile_store<false>(xt, Wskip, nullptr, sout, row0, n0, lane, nrows);
}

// ---------------------------------------------------------------------------
// Kernel 2: edge phase. One wave32 per edge, lane owns 4 consecutive channels
// (float4).  gate = relu(k[dst]+q[src]); agg[dst] += gate * v[src].
// Scatter uses inline-asm global_atomic_add_f32 (no-return, STOREcnt) to
// guarantee single-instruction fp32 atomics (no CAS expansion).
// ---------------------------------------------------------------------------
__device__ __forceinline__ void atomic_add_f32_asm(float* p, float val) {
    asm volatile("global_atomic_add_f32 %0, %1, off"
                 :: "v"(p), "v"(val) : "memory");
}

__global__ __launch_bounds__(256) void edge_kernel(
    const int* __restrict__ ei,
    const float* __restrict__ k, const float* __restrict__ q,
    const float* __restrict__ v, float* __restrict__ agg, int nedges)
{
    const int lane   = threadIdx.x & 31;
    const int waveId = (int)((blockIdx.x * blockDim.x + threadIdx.x) >> 5);
    const int nWaves = (int)((gridDim.x * blockDim.x) >> 5);
    const int c = lane * 4;

    for (int e = waveId; e < nedges; e += nWaves) {
        const int s = ei[e];            // src (row 0 of edge_index)
        const int d = ei[nedges + e];   // dst (row 1)
        const float4 kd = *(const float4*)(k + (size_t)d * DD + c);
        const float4 qs = *(const float4*)(q + (size_t)s * DD + c);
        const float4 vs = *(const float4*)(v + (size_t)s * DD + c);
        float4 g;
        g.x = fmaxf(kd.x + qs.x, 0.f) * vs.x;
        g.y = fmaxf(kd.y + qs.y, 0.f) * vs.y;
        g.z = fmaxf(kd.z + qs.z, 0.f) * vs.z;
        g.w = fmaxf(kd.w + qs.w, 0.f) * vs.w;
        float* ap = agg + (size_t)d * DD + c;
        atomic_add_f32_asm(ap + 0, g.x);
        atomic_add_f32_asm(ap + 1, g.y);
        atomic_add_f32_asm(ap + 2, g.z);
        atomic_add_f32_asm(ap + 3, g.w);
    }
}

// ---------------------------------------------------------------------------
// Kernel 3: epilogue.  x2 = leaky_relu(agg + skip + b_conv) staged in LDS,
// then out = x2 @ Wlin + blin via WMMA.
// ---------------------------------------------------------------------------
__global__ __launch_bounds__(256) void out_kernel(
    const float* __restrict__ agg, const float* __restrict__ skip,
    const float* __restrict__ bconv, const float* __restrict__ Wlin,
    const float* __restrict__ blin, float* __restrict__ out, int nrows)
{
    __shared__ float xt[16 * LDSX];
    const int row0 = blockIdx.x * 16;
    const int tid  = threadIdx.x;

    const bool full = (row0 + 16 <= nrows);   // uniform over the block
    for (int i = tid; i < 512; i += 256) {
        const int r = (i * 4) / DD;
        const int c = (i * 4) % DD;
        float4 t = make_float4(0.f, 0.f, 0.f, 0.f);
        if (full || row0 + r < nrows) {
            const size_t base = (size_t)(row0 + r) * DD + c;
            const float4 a = *(const float4*)(agg + base);
            const float4 s = *(const float4*)(skip + base);
            const float4 b = *(const float4*)(bconv + c);
            t.x = a.x + s.x + b.x;  t.y = a.y + s.y + b.y;
            t.z = a.z + s.z + b.z;  t.w = a.w + s.w + b.w;
            t.x = t.x >= 0.f ? t.x : NEG_SLOPE * t.x;
            t.y = t.y >= 0.f ? t.y : NEG_SLOPE * t.y;
            t.z = t.z >= 0.f ? t.z : NEG_SLOPE * t.z;
            t.w = t.w >= 0.f ? t.w : NEG_SLOPE * t.w;
        }
        float* p = &xt[r * LDSX + c];
        p[0] = t.x; p[1] = t.y; p[2] = t.z; p[3] = t.w;
    }
    __syncthreads();

    const int wave = tid >> 5;
    const int lane = tid & 31;
    gemm_tile_store<true>(xt, Wlin, blin, out, row0, wave * 16, lane, nrows);
}

// ---------------------------------------------------------------------------
extern "C" void kernel_launch(void* const* d_in, const int* in_sizes, int n_in,
                              void* d_out, int out_size, void* d_ws, size_t ws_size,
                              hipStream_t stream)
{
    const float* x     = (const float*)d_in[0];
    const int*   ei    = (const int*)  d_in[1];
    const float* Wk    = (const float*)d_in[2];
    const float* bk    = (const float*)d_in[3];
    const float* Wq    = (const float*)d_in[4];
    const float* bq    = (const float*)d_in[5];
    const float* Wv    = (const float*)d_in[6];
    const float* bv    = (const float*)d_in[7];
    const float* Wskip = (const float*)d_in[8];
    const float* bconv = (const float*)d_in[9];
    const float* Wlin  = (const float*)d_in[10];
    const float* blin  = (const float*)d_in[11];

    const int N_ = in_sizes[0] / DD;     // 50000
    const int E_ = in_sizes[1] / 2;      // 800000

    const size_t ND = (size_t)N_ * DD;
    float* kbuf = (float*)d_ws;
    float* qbuf = kbuf + ND;
    float* vbuf = qbuf + ND;
    float* sbuf = vbuf + ND;
    float* abuf = sbuf + ND;

    hipMemsetAsync(abuf, 0, ND * sizeof(float), stream);

    const int nblocks = (N_ + 15) / 16;  // 3125
    node_proj_kernel<<<nblocks, 256, 0, stream>>>(
        x, Wk, bk, Wq, bq, Wv, bv, Wskip, kbuf, qbuf, vbuf, sbuf, N_);

    edge_kernel<<<2048, 256, 0, stream>>>(ei, kbuf, qbuf, vbuf, abuf, E_);

    out_kernel<<<nblocks, 256, 0, stream>>>(
        abuf, sbuf, bconv, Wlin, blin, (float*)d_out, N_);
}